// RankDPOLoss_28157805592779
// MI455X (gfx1250) — compile-verified
//
#include <hip/hip_runtime.h>
#include <hip/hip_bf16.h>

typedef float v2f __attribute__((ext_vector_type(2)));
typedef float v8f __attribute__((ext_vector_type(8)));

// ---------------------------------------------------------------------------
// Wave32 full-precision sum reduction via one V_WMMA_F32_16X16X4_F32.
// A (16x4 f32, 2 VGPRs/lane): lanes 0-15 -> A[m][0],A[m][1]; lanes 16-31 ->
// A[m][2],A[m][3] (m = lane%16).  We put the lane partial in the first slot
// and 0 in the second, B = all-ones, C = 0.  Then D[m][n] = p[m] + p[m+16]
// for every n.  D layout: lane L holds D[v][L%16] (v=0..7 for L<16, rows
// 8..15 for L>=16), so summing the 8 D components gives:
//   lanes 0-15 : sum_{m=0..7}  (p[m]+p[m+16])
//   lanes 16-31: sum_{m=8..15} (p[m]+p[m+16])
// readlane(0) + readlane(16) == sum of all 32 partials, all in f32.
// Requires EXEC == all ones (call only from fully-converged code).
// ---------------------------------------------------------------------------
__device__ inline float wave_reduce_sum_wmma(float x) {
    v2f a; a.x = x;    a.y = 0.0f;
    v2f b; b.x = 1.0f; b.y = 1.0f;
    v8f c = {};
    v8f d = __builtin_amdgcn_wmma_f32_16x16x4_f32(
        /*neg_a=*/false, a, /*neg_b=*/false, b,
        /*c_mod=*/(short)0, c, /*reuse_a=*/false, /*reuse_b=*/false);
    float s = d[0] + d[1] + d[2] + d[3] + d[4] + d[5] + d[6] + d[7];
    float lo = __int_as_float(__builtin_amdgcn_readlane(__float_as_int(s), 0));
    float hi = __int_as_float(__builtin_amdgcn_readlane(__float_as_int(s), 16));
    return lo + hi;
}

__device__ inline float log_sigmoid_stable(float z) {
    // jax.nn.log_sigmoid(z) = min(z,0) - log1p(exp(-|z|))
    return fminf(z, 0.0f) - log1pf(__expf(-fabsf(z)) * 0.0f + expf(-fabsf(z)));
}

// ---------------------------------------------------------------------------
// Kernel 1: per-row pairwise loss sum.  256 threads = 4 rows/block, 64
// threads (2 waves) per row, lane-per-item.  Row staged in LDS.
// ---------------------------------------------------------------------------
__global__ __launch_bounds__(256)
void rankdpo_row_kernel(const float* __restrict__ S,   // policy_logps [B,64]
                        const float* __restrict__ R,   // reward_scores [B,64]
                        float* __restrict__ row_sums,  // [B]
                        int B) {
    constexpr int K = 64;
    __shared__ float sr[4][K];
    __shared__ float ss[4][K];
    __shared__ float sg[4][K];
    __shared__ float sd[4][K];
    __shared__ float wsum[8];

    const int tid  = threadIdx.x;
    const int sub  = tid >> 6;        // row within block: 0..3
    const int i    = tid & 63;        // item owned by this thread
    int row = blockIdx.x * 4 + sub;
    if (row >= B) row = B - 1;        // clamp: duplicate work, same value written

    const float ri = R[row * K + i];
    const float si = S[row * K + i];
    sr[sub][i] = ri;
    ss[sub][i] = si;
    __syncthreads();

    // rank of item i under stable descending sort of rewards
    int cnt = 0;
#pragma unroll 8
    for (int j = 0; j < K; ++j) {
        const float rj = sr[sub][j];
        cnt += ((rj > ri) || (rj == ri && j < i)) ? 1 : 0;
    }
    const float di = 1.0f / log1pf((float)(cnt + 1));
    const float gi = 2.0f * ri - 1.0f;
    sg[sub][i] = gi;
    sd[sub][i] = di;
    __syncthreads();

    // pairwise terms: this thread covers (i, j) for j < i (strict lower tri)
    float acc = 0.0f;
#pragma unroll 4
    for (int j = 0; j < K; ++j) {
        const float gj = sg[sub][j];
        const float dj = sd[sub][j];
        const float sj = ss[sub][j];
        const float delta = fabsf(gi - gj) * fabsf(di - dj);
        const float z  = sj - si;                                   // -BETA*(s_i - s_j)
        const float ls = fminf(z, 0.0f) - log1pf(expf(-fabsf(z)));  // log_sigmoid(z)
        acc += (j < i) ? delta * ls : 0.0f;
    }

    // wave32 sum via WMMA (all lanes converged here), then combine the two
    // waves of this row through LDS.
    const float wv = wave_reduce_sum_wmma(acc);
    const int wave = tid >> 5;
    if ((tid & 31) == 0) wsum[wave] = wv;
    __syncthreads();
    if (i == 0) row_sums[row] = wsum[sub * 2] + wsum[sub * 2 + 1];
}

// ---------------------------------------------------------------------------
// Kernel 2: deterministic final reduction of B row sums (single block).
// ---------------------------------------------------------------------------
__global__ __launch_bounds__(256)
void rankdpo_reduce_kernel(const float* __restrict__ row_sums,
                           float* __restrict__ out,
                           int B, float neg_inv_count) {
    __shared__ float wsum[8];
    const int tid = threadIdx.x;
    float acc = 0.0f;
    for (int idx = tid; idx < B; idx += 256) acc += row_sums[idx];
    const float wv = wave_reduce_sum_wmma(acc);
    if ((tid & 31) == 0) wsum[tid >> 5] = wv;
    __syncthreads();
    if (tid == 0) {
        float t = 0.0f;
#pragma unroll
        for (int k = 0; k < 8; ++k) t += wsum[k];
        out[0] = t * neg_inv_count;   // -sum / count
    }
}

extern "C" void kernel_launch(void* const* d_in, const int* in_sizes, int n_in,
                              void* d_out, int out_size, void* d_ws, size_t ws_size,
                              hipStream_t stream) {
    constexpr int K = 64;
    const float* policy_logps  = (const float*)d_in[0];
    const float* reward_scores = (const float*)d_in[1];
    float* out = (float*)d_out;
    float* ws  = (float*)d_ws;   // B floats of scratch (32 KB for B=8192)

    const int B = in_sizes[0] / K;
    const long long pairs = (long long)K * (K - 1) / 2;            // 2016
    const float neg_inv_count = (float)(-1.0 / ((double)B * (double)pairs));

    rankdpo_row_kernel<<<(B + 3) / 4, 256, 0, stream>>>(
        policy_logps, reward_scores, ws, B);
    rankdpo_reduce_kernel<<<1, 256, 0, stream>>>(ws, out, B, neg_inv_count);
}